// GraphFP_80367428043545
// MI455X (gfx1250) — compile-verified
//
#include <hip/hip_runtime.h>
#include <hip/hip_bf16.h>

// Problem constants
#define BB    128
#define NN    128
#define INNER 40
#define FF    41
#define OUTD  512
#define DEPTH 2

typedef float v2f __attribute__((ext_vector_type(2)));
typedef float v8f __attribute__((ext_vector_type(8)));

// ---------------------------------------------------------------------------
// Kernel 0: init output (accumulated atomically later) and s = 1.0
// ---------------------------------------------------------------------------
__global__ void gfp_init_kernel(float* __restrict__ out, float* __restrict__ s) {
    int tid = blockIdx.x * blockDim.x + threadIdx.x;
    if (tid < BB * OUTD) out[tid] = 0.0f;
    if (tid < BB * NN)   s[tid]   = 1.0f;
}

// ---------------------------------------------------------------------------
// Kernel 1: single bandwidth-bound pass over g.
// One workgroup per (b, i) row: reads g[b, i, :, :] (128*41 floats, coalesced
// float4), producing:
//   pre[b,i,c]   = sum_j g[b,i,j,c]          (c < 40)   via LDS f32 atomics
//   s[b,j]      += g[b,i,j,40]                           via global f32 atomics
//   attrs[b,i,c] = g[b,i,i,c]                (c < 40)
// ---------------------------------------------------------------------------
__global__ __launch_bounds__(256) void gfp_reduce_kernel(
    const float* __restrict__ g,
    float* __restrict__ pre, float* __restrict__ attrs, float* __restrict__ s) {
    __shared__ float preL[INNER];
    const int tid = threadIdx.x;
    if (tid < INNER) preL[tid] = 0.0f;
    __syncthreads();

    const int row = blockIdx.x;        // b*NN + i
    const int b   = row >> 7;
    const int i   = row & (NN - 1);
    const float* base = g + (size_t)row * (NN * FF);
    const float4* base4 = (const float4*)base;   // 128*41 = 5248 floats = 1312 float4

    for (int q = tid; q < (NN * FF) / 4; q += 256) {
        float4 v = base4[q];
        int e0 = q * 4;
#pragma unroll
        for (int c = 0; c < 4; ++c) {
            int e = e0 + c;
            int j = e / FF;
            int f = e - j * FF;
            float val = (c == 0) ? v.x : (c == 1) ? v.y : (c == 2) ? v.z : v.w;
            if (f < INNER) {
                atomicAdd(&preL[f], val);
                if (j == i) attrs[(size_t)row * INNER + f] = val;
            } else {
                atomicAdd(&s[b * NN + j], val);   // bonds column-sum into s[b, j]
            }
        }
    }
    __syncthreads();
    if (tid < INNER) pre[(size_t)row * INNER + tid] = preL[tid];
}

// ---------------------------------------------------------------------------
// Kernel 2: per (batch, 16-row tile) depth recurrence + WMMA logits GEMM
// + softmax + column-sum accumulation into out[b, :].
// 128 threads = 4 waves; wave w owns logits columns [128*w, 128*w+128).
// ---------------------------------------------------------------------------
__global__ __launch_bounds__(128) void gfp_fp_kernel(
    const float* __restrict__ pre_g, const float* __restrict__ attrs_g,
    const float* __restrict__ s_g,
    const float* __restrict__ W_inner, const float* __restrict__ b_inner,
    const float* __restrict__ W_output, const float* __restrict__ b_output,
    float* __restrict__ out) {

    __shared__ float attrsL[16][42];   // padded stride 42: no LDS bank conflicts
    __shared__ float preL[16][42];
    __shared__ float newcL[16][42];
    __shared__ float sv[16];
    __shared__ float pred[4][16];      // per-wave row partial (max / sum)
    __shared__ float comb[16];         // combined row stat

    const int tid  = threadIdx.x;
    const int wave = tid >> 5;
    const int lane = tid & 31;
    const int ml   = lane & 15;        // M (A) / N (B,C) index within tile
    const int hi   = lane >> 4;        // half-wave select

    const int bidx    = blockIdx.x >> 3;       // batch
    const int mt      = blockIdx.x & 7;        // 16-row tile within batch
    const int rowbase = bidx * NN + mt * 16;

    // Load row-tile state
    for (int q = tid; q < 16 * INNER; q += 128) {
        int r = q / INNER, c = q - r * INNER;
        attrsL[r][c] = attrs_g[(size_t)(rowbase + r) * INNER + c];
        preL[r][c]   = pre_g[(size_t)(rowbase + r) * INNER + c];
    }
    if (tid < 16) sv[tid] = s_g[rowbase + tid];
    __syncthreads();

    for (int d = 0; d <= DEPTH; ++d) {
        if (d > 0) {
            // new_core = preL @ W_inner[d] + b_inner[d]   (16x40 @ 40x40, scalar)
            const float* Wi = W_inner + d * INNER * INNER;
            const float* bi = b_inner + d * INNER;
            for (int q = tid; q < 16 * INNER; q += 128) {
                int r = q / INNER, c = q - r * INNER;
                float acc = bi[c];
                for (int k = 0; k < INNER; ++k) acc += preL[r][k] * Wi[k * INNER + c];
                newcL[r][c] = acc;
            }
            __syncthreads();
            // delta update: pre += s * (new - attrs); attrs = new
            for (int q = tid; q < 16 * INNER; q += 128) {
                int r = q / INNER, c = q - r * INNER;
                float nv = newcL[r][c];
                preL[r][c] += sv[r] * (nv - attrsL[r][c]);
                attrsL[r][c] = nv;
            }
            __syncthreads();
        }

        // ---- logits = attrsL @ W_output[d] + b_output[d], per-wave 16x128 ----
        const float* Wo  = W_output + d * INNER * OUTD;
        const float* bo  = b_output + d * OUTD;
        const int    ncb = wave * 128;

        v8f acc[8];
#pragma unroll
        for (int t = 0; t < 8; ++t)
#pragma unroll
            for (int r = 0; r < 8; ++r) acc[t][r] = 0.0f;

        for (int k0 = 0; k0 < INNER; k0 += 4) {
            // f32 A fragment (16x4): lane<16 -> K = k0,k0+1 ; lane>=16 -> k0+2,k0+3
            v2f a;
            a.x = attrsL[ml][k0 + 2 * hi];
            a.y = attrsL[ml][k0 + 2 * hi + 1];
#pragma unroll
            for (int t = 0; t < 8; ++t) {
                int col = ncb + 16 * t + ml;
                v2f bb;
                bb.x = Wo[(k0 + 2 * hi) * OUTD + col];
                bb.y = Wo[(k0 + 2 * hi + 1) * OUTD + col];
                acc[t] = __builtin_amdgcn_wmma_f32_16x16x4_f32(
                    false, a, false, bb, (short)0, acc[t], false, false);
            }
        }

        // bias (column-wise)
#pragma unroll
        for (int t = 0; t < 8; ++t) {
            float bv = bo[ncb + 16 * t + ml];
#pragma unroll
            for (int r = 0; r < 8; ++r) acc[t][r] += bv;
        }

        // ---- row max over full 512 columns ----
        float rm[8];
#pragma unroll
        for (int r = 0; r < 8; ++r) {
            float m = acc[0][r];
#pragma unroll
            for (int t = 1; t < 8; ++t) m = fmaxf(m, acc[t][r]);
#pragma unroll
            for (int off = 1; off <= 8; off <<= 1)
                m = fmaxf(m, __shfl_xor(m, off, 32));
            rm[r] = m;
        }
        if (ml == 0) {
#pragma unroll
            for (int r = 0; r < 8; ++r) pred[wave][r + 8 * hi] = rm[r];
        }
        __syncthreads();
        if (tid < 16)
            comb[tid] = fmaxf(fmaxf(pred[0][tid], pred[1][tid]),
                              fmaxf(pred[2][tid], pred[3][tid]));
        __syncthreads();

        float cm[8];
#pragma unroll
        for (int r = 0; r < 8; ++r) cm[r] = comb[r + 8 * hi];

        // ---- exponentiate + row sums ----
        float rs[8];
#pragma unroll
        for (int r = 0; r < 8; ++r) rs[r] = 0.0f;
#pragma unroll
        for (int t = 0; t < 8; ++t)
#pragma unroll
            for (int r = 0; r < 8; ++r) {
                float e = __expf(acc[t][r] - cm[r]);
                acc[t][r] = e;
                rs[r] += e;
            }
#pragma unroll
        for (int r = 0; r < 8; ++r) {
#pragma unroll
            for (int off = 1; off <= 8; off <<= 1)
                rs[r] += __shfl_xor(rs[r], off, 32);
        }
        if (ml == 0) {
#pragma unroll
            for (int r = 0; r < 8; ++r) pred[wave][r + 8 * hi] = rs[r];
        }
        __syncthreads();
        if (tid < 16)
            comb[tid] = pred[0][tid] + pred[1][tid] + pred[2][tid] + pred[3][tid];
        __syncthreads();

        float ri[8];
#pragma unroll
        for (int r = 0; r < 8; ++r) ri[r] = 1.0f / comb[r + 8 * hi];

        // ---- column sums of softmax rows -> accumulate into out[b, :] ----
#pragma unroll
        for (int t = 0; t < 8; ++t) {
            float cv = 0.0f;
#pragma unroll
            for (int r = 0; r < 8; ++r) cv += acc[t][r] * ri[r];
            cv += __shfl_xor(cv, 16, 32);      // combine the two 8-row halves
            if (hi == 0)
                atomicAdd(&out[bidx * OUTD + ncb + 16 * t + ml], cv);
        }
        __syncthreads();   // protect comb/pred/attrsL reuse next depth
    }
}

// ---------------------------------------------------------------------------
extern "C" void kernel_launch(void* const* d_in, const int* in_sizes, int n_in,
                              void* d_out, int out_size, void* d_ws, size_t ws_size,
                              hipStream_t stream) {
    const float* g  = (const float*)d_in[0];   // (B,N,N,F)
    const float* Wi = (const float*)d_in[1];   // (3,40,40)
    const float* bi = (const float*)d_in[2];   // (3,1,40)
    const float* Wo = (const float*)d_in[3];   // (3,40,512)
    const float* bo = (const float*)d_in[4];   // (3,1,512)
    float* out = (float*)d_out;                // (B,512)

    float* pre   = (float*)d_ws;               // B*N*40
    float* attrs = pre + (size_t)BB * NN * INNER;
    float* s     = attrs + (size_t)BB * NN * INNER;

    gfp_init_kernel<<<(BB * OUTD + 255) / 256, 256, 0, stream>>>(out, s);
    gfp_reduce_kernel<<<BB * NN, 256, 0, stream>>>(g, pre, attrs, s);
    gfp_fp_kernel<<<BB * 8, 128, 0, stream>>>(pre, attrs, s, Wi, bi, Wo, bo, out);
}